// DEAlignment_56831007261074
// MI455X (gfx1250) — compile-verified
//
#include <hip/hip_runtime.h>
#include <math.h>

#define BATCH      32
#define NPTS       16384
#define SUB        128
#define P_PART     50
#define N_ITERS    30
#define ROT_RANGE  0.7853981633974483f   /* radians(45) */
#define TR_RANGE   1.0f
#define F_DE       0.8f
#define CR_DE      0.9f
#define FLTMAXBITS 0x7f7fffffu

typedef float v2f __attribute__((ext_vector_type(2)));
typedef float v8f __attribute__((ext_vector_type(8)));

// ---------------- RNG (deterministic, counter-based) ----------------
__device__ __forceinline__ unsigned pcg(unsigned x) {
    x = x * 747796405u + 2891336453u;
    unsigned w = ((x >> ((x >> 28u) + 4u)) ^ x) * 277803737u;
    return (w >> 22u) ^ w;
}
__device__ __forceinline__ float u01(unsigned h) {
    return (float)(h >> 8) * (1.0f / 16777216.0f);
}

// ---------------- axis-angle -> rotation matrix ----------------
// R = c*I + s*K + (1-c) k k^T   (row-major R[k*3+j])
__device__ __forceinline__ void axisangle_to_R(float vx, float vy, float vz, float* R) {
    float th  = sqrtf(vx * vx + vy * vy + vz * vz);
    float inv = 1.0f / fmaxf(th, 1e-8f);
    float kx = vx * inv, ky = vy * inv, kz = vz * inv;
    float s = sinf(th), c = cosf(th), oc = 1.0f - c;
    R[0] = c + oc * kx * kx;  R[1] = oc * kx * ky - s * kz;  R[2] = oc * kx * kz + s * ky;
    R[3] = oc * kx * ky + s * kz;  R[4] = c + oc * ky * ky;  R[5] = oc * ky * kz - s * kx;
    R[6] = oc * kx * kz - s * ky;  R[7] = oc * ky * kz + s * kx;  R[8] = c + oc * kz * kz;
    if (th < 1e-8f) {
        R[0] = 1.f; R[1] = 0.f; R[2] = 0.f;
        R[3] = 0.f; R[4] = 1.f; R[5] = 0.f;
        R[6] = 0.f; R[7] = 0.f; R[8] = 1.f;
    }
}

// ---------------- prep: subsample + |y|^2 + init population ----------------
__global__ void prep_kernel(const float* __restrict__ src, const float* __restrict__ tgt,
                            float* __restrict__ srcsub, float* __restrict__ tgtsub,
                            float* __restrict__ yn, float* __restrict__ pop0) {
    int b = blockIdx.x, tid = threadIdx.x;
    if (tid < SUB) {
        // multiplicative bijections mod 2^14 (odd multipliers)
        int is_ = (tid * 2687 + 911)  & (NPTS - 1);
        int it_ = (tid * 3571 + 1733) & (NPTS - 1);
        const float* s = src + ((size_t)b * NPTS + is_) * 3;
        const float* t = tgt + ((size_t)b * NPTS + it_) * 3;
        ((float4*)srcsub)[b * SUB + tid] = make_float4(s[0], s[1], s[2], 0.f);
        float t0 = t[0], t1 = t[1], t2 = t[2];
        ((float4*)tgtsub)[b * SUB + tid] = make_float4(t0, t1, t2, 0.f);
        yn[b * SUB + tid] = t0 * t0 + t1 * t1 + t2 * t2;
    }
    for (int e = tid; e < P_PART * 6; e += blockDim.x) {
        int p = e / 6, d = e % 6;
        float v = 0.0f;
        if (p != 0) {
            unsigned h = pcg((unsigned)((b * P_PART + p) * 8 + d) * 2654435761u + 0x9E3779B9u);
            float u = u01(h) * 2.0f - 1.0f;
            v = (d < 3) ? u * ROT_RANGE : u * TR_RANGE;
        }
        pop0[(b * P_PART + p) * 6 + d] = v;
    }
}

// ---------------- DE step + WMMA chamfer evaluation ----------------
// mode 0: evaluate pop_in -> fit_out (initial fitness)
// mode 1: build trial (DE/rand/1/bin), evaluate, select -> pop_out/fit_out
__global__ void de_kernel(int mode, int it,
                          const float* __restrict__ pop_in, const float* __restrict__ fit_in,
                          float* __restrict__ pop_out, float* __restrict__ fit_out,
                          const float* __restrict__ srcsub, const float* __restrict__ tgtsub,
                          const float* __restrict__ ynq) {
    int bp = blockIdx.x;
    int b = bp / P_PART, p = bp % P_PART;
    int tid = threadIdx.x;

    __shared__ float trial[8];
    // k-interleaved point storage: [point][k], k=0..2 coords, k=3 == 0 (K-pad for WMMA)
    __shared__ float xs4[SUB * 4];
    __shared__ float tt4[SUB * 4];
    __shared__ float xn[SUB], yn[SUB];
    __shared__ unsigned mrow[SUB], mcol[SUB];
    __shared__ float rbuf[SUB];

    // --- build trial vector (thread 0) ---
    if (tid == 0) {
        const float* self = pop_in + (size_t)bp * 6;
        if (mode == 0) {
            for (int d = 0; d < 6; ++d) trial[d] = self[d];
        } else {
            unsigned h = pcg(pcg(pcg(0xC0FFEEu ^ (unsigned)it * 1664525u)
                                 ^ (unsigned)b * 22695477u) ^ (unsigned)p * 747796405u);
            auto next = [&]() { h = pcg(h); return h; };
            int r1 = (int)(next() % P_PART); while (r1 == p) r1 = (r1 + 1) % P_PART;
            int r2 = (int)(next() % P_PART); while (r2 == p || r2 == r1) r2 = (r2 + 1) % P_PART;
            int r3 = (int)(next() % P_PART); while (r3 == p || r3 == r1 || r3 == r2) r3 = (r3 + 1) % P_PART;
            int jrand = (int)(next() % 6u);
            const float* x1 = pop_in + (size_t)(b * P_PART + r1) * 6;
            const float* x2 = pop_in + (size_t)(b * P_PART + r2) * 6;
            const float* x3 = pop_in + (size_t)(b * P_PART + r3) * 6;
            for (int d = 0; d < 6; ++d) {
                float u = u01(next());
                bool cm = (u < CR_DE) || (d == jrand);
                float m = x1[d] + F_DE * (x2[d] - x3[d]);
                float lim = (d < 3) ? ROT_RANGE : TR_RANGE;
                m = fminf(fmaxf(m, -lim), lim);
                trial[d] = cm ? m : self[d];
            }
        }
    }
    __syncthreads();

    // --- transform subsampled source; stage target; init reductions ---
    float R[9];
    axisangle_to_R(trial[0], trial[1], trial[2], R);
    float trx = trial[3], try_ = trial[4], trz = trial[5];
    if (tid < SUB) {
        float4 sv = ((const float4*)srcsub)[b * SUB + tid];          // global_load_b128
        float x0 = R[0] * sv.x + R[1] * sv.y + R[2] * sv.z + trx;
        float x1 = R[3] * sv.x + R[4] * sv.y + R[5] * sv.z + try_;
        float x2 = R[6] * sv.x + R[7] * sv.y + R[8] * sv.z + trz;
        ((float4*)xs4)[tid] = make_float4(x0, x1, x2, 0.f);          // ds_store_b128
        xn[tid] = x0 * x0 + x1 * x1 + x2 * x2;
        mrow[tid] = FLTMAXBITS; mcol[tid] = FLTMAXBITS;
    } else {
        int j = tid - SUB;
        float4 tv = ((const float4*)tgtsub)[b * SUB + j];            // global_load_b128
        ((float4*)tt4)[j] = tv;                                      // ds_store_b128 (w==0)
        yn[j] = ynq[b * SUB + j];
    }
    __syncthreads();

    // --- 128x128 distance matrix via V_WMMA_F32_16X16X4_F32 (K=3 padded to 4) ---
    int wave = tid >> 5, lane = tid & 31;
    int l15 = lane & 15;
    bool hi = lane >= 16;
    int koff = hi ? 2 : 0;              // lo lanes: K0,K1  |  hi lanes: K2,K3(=0)
    int rowbase = wave * 16;            // wave owns row tile ti = wave

    // A frag: one branch-free ds_load_b64 (k-interleaved, zero-padded)
    v2f a = *(const v2f*)&xs4[(rowbase + l15) * 4 + koff];

    // row norms for this lane's 8 rows: two ds_load_b128
    float xnv[8];
    {
        const float4* xp = (const float4*)&xn[rowbase + (hi ? 8 : 0)];
        float4 xa = xp[0], xb = xp[1];
        xnv[0] = xa.x; xnv[1] = xa.y; xnv[2] = xa.z; xnv[3] = xa.w;
        xnv[4] = xb.x; xnv[5] = xb.y; xnv[6] = xb.z; xnv[7] = xb.w;
    }

    float rmin[8];
#pragma unroll
    for (int r = 0; r < 8; ++r) rmin[r] = 3.402823466e38f;

#pragma unroll
    for (int tj = 0; tj < 8; ++tj) {
        int cc = tj * 16 + l15;
        v2f bvec = *(const v2f*)&tt4[cc * 4 + koff];   // ds_load_b64, branch-free
        v8f c = {};
        c = __builtin_amdgcn_wmma_f32_16x16x4_f32(false, a, false, bvec,
                                                  (short)0, c, false, false);
        float ynm = yn[cc];
        float cmin = 3.402823466e38f;
#pragma unroll
        for (int r = 0; r < 8; ++r) {
            float d = xnv[r] + ynm - 2.0f * c[r];
            d = fmaxf(d, 0.0f);
            rmin[r] = fminf(rmin[r], d);
            cmin = fminf(cmin, d);
        }
        atomicMin(&mcol[cc], __float_as_uint(cmin));   // d>=0: uint order == float order
    }
#pragma unroll
    for (int r = 0; r < 8; ++r)
        atomicMin(&mrow[rowbase + r + (hi ? 8 : 0)], __float_as_uint(rmin[r]));
    __syncthreads();

    // --- deterministic tree reduction: fitness = mean(rowmin) + mean(colmin) ---
    if (tid < SUB)
        rbuf[tid] = __uint_as_float(mrow[tid]) + __uint_as_float(mcol[tid]);
    __syncthreads();
    for (int s = 64; s > 0; s >>= 1) {
        if (tid < s) rbuf[tid] += rbuf[tid + s];
        __syncthreads();
    }
    float tfit = rbuf[0] * (1.0f / (float)SUB);

    // --- selection ---
    if (mode == 0) {
        if (tid == 0) fit_out[bp] = tfit;
    } else {
        float oldfit = fit_in[bp];
        bool imp = tfit < oldfit;
        if (tid < 6) pop_out[(size_t)bp * 6 + tid] = imp ? trial[tid] : pop_in[(size_t)bp * 6 + tid];
        if (tid == 0) fit_out[bp] = imp ? tfit : oldfit;
    }
}

// ---------------- argmin over particles; emit R_best, t_best ----------------
__global__ void finalize_kernel(const float* __restrict__ pop, const float* __restrict__ fit,
                                float* __restrict__ out) {
    int b = blockIdx.x, tid = threadIdx.x;   // 64 threads
    __shared__ float bfv[64];
    __shared__ int   bfi[64];
    float v = 3.402823466e38f; int idx = P_PART;
    for (int p = tid; p < P_PART; p += 64) {
        float f = fit[b * P_PART + p];
        if (f < v || (f == v && p < idx)) { v = f; idx = p; }
    }
    bfv[tid] = v; bfi[tid] = idx;
    __syncthreads();
    for (int s = 32; s > 0; s >>= 1) {
        if (tid < s) {
            float f2 = bfv[tid + s]; int i2 = bfi[tid + s];
            if (f2 < bfv[tid] || (f2 == bfv[tid] && i2 < bfi[tid])) { bfv[tid] = f2; bfi[tid] = i2; }
        }
        __syncthreads();
    }
    if (tid == 0) {
        const float* pose = pop + (size_t)(b * P_PART + bfi[0]) * 6;
        float R[9];
        axisangle_to_R(pose[0], pose[1], pose[2], R);
        for (int k = 0; k < 9; ++k) out[b * 9 + k] = R[k];
        out[BATCH * 9 + b * 3 + 0] = pose[3];
        out[BATCH * 9 + b * 3 + 1] = pose[4];
        out[BATCH * 9 + b * 3 + 2] = pose[5];
    }
}

// ---------------- full-resolution alignment (b128 vectorized) ----------------
__global__ void aligned_kernel(const float* __restrict__ src, float* __restrict__ out) {
    int tid4 = blockIdx.x * blockDim.x + threadIdx.x;  // 4 points / thread
    int b = blockIdx.x >> 4;                           // 1024 points per block, 16 blocks/batch
    const float* Rp = out + (size_t)b * 9;
    const float* tp = out + BATCH * 9 + (size_t)b * 3;
    float R0 = Rp[0], R1 = Rp[1], R2 = Rp[2];
    float R3 = Rp[3], R4 = Rp[4], R5 = Rp[5];
    float R6 = Rp[6], R7 = Rp[7], R8 = Rp[8];
    float t0 = tp[0], t1 = tp[1], t2 = tp[2];

    const float4* sp = (const float4*)src + (size_t)tid4 * 3;
    float4 s0 = sp[0], s1 = sp[1], s2 = sp[2];

    float o[12];
#define XF(sx, sy, sz, i)                                   \
    o[(i)*3 + 0] = R0 * (sx) + R1 * (sy) + R2 * (sz) + t0;  \
    o[(i)*3 + 1] = R3 * (sx) + R4 * (sy) + R5 * (sz) + t1;  \
    o[(i)*3 + 2] = R6 * (sx) + R7 * (sy) + R8 * (sz) + t2;
    XF(s0.x, s0.y, s0.z, 0)
    XF(s0.w, s1.x, s1.y, 1)
    XF(s1.z, s1.w, s2.x, 2)
    XF(s2.y, s2.z, s2.w, 3)
#undef XF

    float4* op = (float4*)(out + BATCH * 12) + (size_t)tid4 * 3;
    op[0] = make_float4(o[0], o[1], o[2], o[3]);
    op[1] = make_float4(o[4], o[5], o[6], o[7]);
    op[2] = make_float4(o[8], o[9], o[10], o[11]);
}

// ---------------- host orchestration ----------------
extern "C" void kernel_launch(void* const* d_in, const int* in_sizes, int n_in,
                              void* d_out, int out_size, void* d_ws, size_t ws_size,
                              hipStream_t stream) {
    const float* src = (const float*)d_in[0];
    const float* tgt = (const float*)d_in[1];
    float* out = (float*)d_out;
    float* ws  = (float*)d_ws;

    float* srcsub = ws;                                  // 32*128*4
    float* tgtsub = srcsub + BATCH * SUB * 4;            // 32*128*4
    float* yn     = tgtsub + BATCH * SUB * 4;            // 32*128
    float* pop0   = yn + BATCH * SUB;                    // 32*50*6
    float* pop1   = pop0 + BATCH * P_PART * 6;
    float* fit0   = pop1 + BATCH * P_PART * 6;           // 32*50
    float* fit1   = fit0 + BATCH * P_PART;
    float* pops[2] = { pop0, pop1 };
    float* fits[2] = { fit0, fit1 };

    prep_kernel<<<BATCH, 256, 0, stream>>>(src, tgt, srcsub, tgtsub, yn, pop0);

    // initial fitness
    de_kernel<<<BATCH * P_PART, 256, 0, stream>>>(0, 0, pop0, fit0, pop1, fit0,
                                                  srcsub, tgtsub, yn);
    int cur = 0;
    for (int it = 0; it < N_ITERS; ++it) {
        de_kernel<<<BATCH * P_PART, 256, 0, stream>>>(1, it,
                                                      pops[cur], fits[cur],
                                                      pops[cur ^ 1], fits[cur ^ 1],
                                                      srcsub, tgtsub, yn);
        cur ^= 1;
    }

    finalize_kernel<<<BATCH, 64, 0, stream>>>(pops[cur], fits[cur], out);

    int blocks = (BATCH * NPTS / 4) / 256;   // 512
    aligned_kernel<<<blocks, 256, 0, stream>>>(src, out);
}